// EvolveGCNH_61924838473855
// MI455X (gfx1250) — compile-verified
//
#include <hip/hip_runtime.h>
#include <hip/hip_bf16.h>
#include <float.h>

// ---------------------------------------------------------------------------
// EvolveGCN-H forward for MI455X (gfx1250, wave32).
//   N=50000 nodes, C=128 channels, E=800000 edges (sizes derived at launch).
// GEMMs use V_WMMA_F32_16X16X4_F32 (fp32 matrix pipe, one wave per 16x16 tile).
// ---------------------------------------------------------------------------

typedef float v2f __attribute__((ext_vector_type(2)));
typedef float v8f __attribute__((ext_vector_type(8)));

#define CCH 128            // channel count
#define WAVE 32

// ---------------------------------------------------------------------------
// D[M,N] = A[M,K] @ W[N,K]^T   (i.e. B[k][n] = W[n][k]), all fp32 row-major.
// One wave computes one 16x16 tile. M, N, K are multiples of 16/4.
// A-matrix VGPR layout (16x4 f32): lanes 0-15 -> K=0,1 ; lanes 16-31 -> K=2,3
// B-matrix (4x16 f32): mirrored (lanes hold column n, khalf by lane>>4).
// C/D (16x16 f32): VGPR r holds row (r + 8*(lane>>4)), col = lane&15.
// ---------------------------------------------------------------------------
__global__ void wmma_gemm_nt_f32(const float* __restrict__ A,
                                 const float* __restrict__ W,
                                 float* __restrict__ D,
                                 int M, int N, int K) {
    const int wavesPerBlock = blockDim.x >> 5;
    const int wave = blockIdx.x * wavesPerBlock + (threadIdx.x >> 5);
    const int lane = threadIdx.x & 31;
    const int ntiles = N >> 4;
    const int tileM = wave / ntiles;
    const int tileN = wave - tileM * ntiles;
    if (tileM * 16 >= M) return;                 // uniform per wave -> EXEC all-1s

    const int mrow  = tileM * 16 + (lane & 15);  // A row this lane feeds
    const int ncol  = tileN * 16 + (lane & 15);  // B col this lane feeds
    const int khalf = (lane >> 4) * 2;           // 0 for lanes 0-15, 2 for 16-31

    const float* arow = A + (size_t)mrow * K + khalf;
    const float* wrow = W + (size_t)ncol * K + khalf;

    v8f acc = {0.f, 0.f, 0.f, 0.f, 0.f, 0.f, 0.f, 0.f};
    for (int k = 0; k < K; k += 4) {
        v2f a = *(const v2f*)(arow + k);
        v2f b = *(const v2f*)(wrow + k);
        acc = __builtin_amdgcn_wmma_f32_16x16x4_f32(
            /*neg_a=*/false, a, /*neg_b=*/false, b,
            /*c_mod=*/(short)0, acc, /*reuse_a=*/false, /*reuse_b=*/false);
    }

    const int col     = tileN * 16 + (lane & 15);
    const int rowBase = tileM * 16 + ((lane >> 4) * 8);
#pragma unroll
    for (int r = 0; r < 8; ++r)
        D[(size_t)(rowBase + r) * N + col] = acc[r];
}

// --------------------------- 1/||p|| (single wave) -------------------------
__global__ void pnorm_kernel(const float* __restrict__ p, float* __restrict__ invnorm) {
    const int lane = threadIdx.x & 31;
    float4 q = ((const float4*)p)[lane];         // 128 floats = 32 x float4
    float s = q.x * q.x + q.y * q.y + q.z * q.z + q.w * q.w;
#pragma unroll
    for (int off = 16; off > 0; off >>= 1) s += __shfl_xor(s, off, 32);
    if (lane == 0) invnorm[0] = 1.0f / sqrtf(s);
}

// ------------------- score = X·p * invnorm (wave per node) -----------------
__global__ void score_kernel(const float* __restrict__ X, const float* __restrict__ p,
                             const float* __restrict__ invnorm,
                             float* __restrict__ score, float* __restrict__ score_copy,
                             int nnodes) {
    const int wave = blockIdx.x * (blockDim.x >> 5) + (threadIdx.x >> 5);
    const int lane = threadIdx.x & 31;
    if (wave >= nnodes) return;
    float4 x = ((const float4*)(X + (size_t)wave * CCH))[lane];
    float4 q = ((const float4*)p)[lane];
    float s = x.x * q.x + x.y * q.y + x.z * q.z + x.w * q.w;
#pragma unroll
    for (int off = 16; off > 0; off >>= 1) s += __shfl_xor(s, off, 32);
    if (lane == 0) {
        float v = s * invnorm[0];
        score[wave] = v;
        score_copy[wave] = v;
    }
}

// ----------- top-128 via 128 deterministic block-wide argmax passes --------
// Ties resolved to the lower index (matches jax.lax.top_k). Destroys sc[].
__global__ void topk_kernel(float* __restrict__ sc, float* __restrict__ topv,
                            int* __restrict__ topi, int nnodes) {
    __shared__ float sval[1024];
    __shared__ int   sidx[1024];
    const int t = threadIdx.x;
    for (int it = 0; it < CCH; ++it) {
        float best = -FLT_MAX;
        int   bi   = 0x7FFFFFFF;
        for (int i = t; i < nnodes; i += 1024) {
            float v = sc[i];
            if (v > best) { best = v; bi = i; }
        }
        sval[t] = best; sidx[t] = bi;
        __syncthreads();
        for (int s = 512; s > 0; s >>= 1) {
            if (t < s) {
                float v2 = sval[t + s]; int i2 = sidx[t + s];
                if (v2 > sval[t] || (v2 == sval[t] && i2 < sidx[t])) {
                    sval[t] = v2; sidx[t] = i2;
                }
            }
            __syncthreads();
        }
        if (t == 0) {
            topv[it] = sval[0];
            topi[it] = sidx[0];
            sc[sidx[0]] = -FLT_MAX;
        }
        __syncthreads();
    }
}

// --------------- X_tilde[i] = X[top_idx[i]] * tanh(top_vals[i]) ------------
__global__ void gather_gate_kernel(const float* __restrict__ X,
                                   const int* __restrict__ topi,
                                   const float* __restrict__ topv,
                                   float* __restrict__ xt) {
    const int row = blockIdx.x;          // 0..127
    const int c   = threadIdx.x;         // 0..127
    const float g = tanhf(topv[row]);
    xt[row * CCH + c] = X[(size_t)topi[row] * CCH + c] * g;
}

// ---------------------- GRU gates -> W_new [128,128] -----------------------
__global__ void gru_gate_kernel(const float* __restrict__ gi, const float* __restrict__ gh,
                                const float* __restrict__ bih, const float* __restrict__ bhh,
                                const float* __restrict__ h,  float* __restrict__ wnew) {
    const int idx = blockIdx.x * blockDim.x + threadIdx.x;   // 0..16383
    const int m = idx >> 7, c = idx & (CCH - 1);
    const int row = m * 3 * CCH;
    float gir = gi[row + c]           + bih[c];
    float giz = gi[row + CCH + c]     + bih[CCH + c];
    float gin = gi[row + 2 * CCH + c] + bih[2 * CCH + c];
    float ghr = gh[row + c]           + bhh[c];
    float ghz = gh[row + CCH + c]     + bhh[CCH + c];
    float ghn = gh[row + 2 * CCH + c] + bhh[2 * CCH + c];
    float r  = 1.0f / (1.0f + __expf(-(gir + ghr)));
    float z  = 1.0f / (1.0f + __expf(-(giz + ghz)));
    float nn = tanhf(gin + r * ghn);
    wnew[idx] = (1.0f - z) * nn + z * h[idx];
}

// ------------------------------- utilities ---------------------------------
__global__ void zero_f32_kernel(float* __restrict__ p, int n) {
    int i = blockIdx.x * blockDim.x + threadIdx.x;
    if (i < n) p[i] = 0.0f;
}

__global__ void init_out_kernel(float* __restrict__ out, const float* __restrict__ bconv, int n) {
    int i = blockIdx.x * blockDim.x + threadIdx.x;
    if (i < n) out[i] = bconv[i & (CCH - 1)];
}

// -------- deg[dst] += w over E edges (self-loop +1 folded into dinv) -------
__global__ void degree_kernel(const long long* __restrict__ ei, const float* __restrict__ w,
                              float* __restrict__ deg, int E) {
    int e = blockIdx.x * blockDim.x + threadIdx.x;
    if (e >= E) return;
    int dst = (int)ei[(size_t)E + e];
    __hip_atomic_fetch_add(&deg[dst], w[e], __ATOMIC_RELAXED, __HIP_MEMORY_SCOPE_AGENT);
}

__global__ void dinv_kernel(const float* __restrict__ deg, float* __restrict__ dinv, int n) {
    int i = blockIdx.x * blockDim.x + threadIdx.x;
    if (i >= n) return;
    float d = deg[i] + 1.0f;   // + self-loop weight 1
    dinv[i] = (d > 0.0f) ? rsqrtf(d) : 0.0f;
}

// ---- out[dst] += Xw[src] * (dinv[src]*w*dinv[dst]) ; one wave per edge ----
// Edges e < E come from edge_index; e in [E, E+N) are the self-loops.
__global__ void scatter_kernel(const long long* __restrict__ ei, const float* __restrict__ w,
                               const float* __restrict__ dinv, const float* __restrict__ Xw,
                               float* __restrict__ out, int E, int nnodes) {
    const int e    = blockIdx.x * (blockDim.x >> 5) + (threadIdx.x >> 5);
    const int lane = threadIdx.x & 31;
    const int total = E + nnodes;
    if (e >= total) return;
    int src, dst; float ww;
    if (e < E) {
        src = (int)ei[e];
        dst = (int)ei[(size_t)E + e];
        ww  = w[e];
    } else {
        src = dst = e - E;
        ww  = 1.0f;
    }
    const float norm = dinv[src] * ww * dinv[dst];
    const float4 x = ((const float4*)(Xw + (size_t)src * CCH))[lane];
    float* o = out + (size_t)dst * CCH + lane * 4;
    __hip_atomic_fetch_add(o + 0, x.x * norm, __ATOMIC_RELAXED, __HIP_MEMORY_SCOPE_AGENT);
    __hip_atomic_fetch_add(o + 1, x.y * norm, __ATOMIC_RELAXED, __HIP_MEMORY_SCOPE_AGENT);
    __hip_atomic_fetch_add(o + 2, x.z * norm, __ATOMIC_RELAXED, __HIP_MEMORY_SCOPE_AGENT);
    __hip_atomic_fetch_add(o + 3, x.w * norm, __ATOMIC_RELAXED, __HIP_MEMORY_SCOPE_AGENT);
}

// ---------------------------------------------------------------------------
extern "C" void kernel_launch(void* const* d_in, const int* in_sizes, int n_in,
                              void* d_out, int out_size, void* d_ws, size_t ws_size,
                              hipStream_t stream) {
    const float*     X     = (const float*)d_in[0];
    const float*     ew    = (const float*)d_in[1];
    const float*     p     = (const float*)d_in[2];
    const float*     W_ih  = (const float*)d_in[3];
    const float*     W_hh  = (const float*)d_in[4];
    const float*     b_ih  = (const float*)d_in[5];
    const float*     b_hh  = (const float*)d_in[6];
    const float*     W_cv  = (const float*)d_in[7];
    const float*     b_cv  = (const float*)d_in[8];
    const long long* ei    = (const long long*)d_in[9];

    const int nnodes = in_sizes[0] / CCH;      // 50000
    const int E      = in_sizes[1];            // 800000

    // ---- workspace carve-up (256B aligned) ----
    char* ws = (char*)d_ws;
    size_t off = 0;
    auto alloc = [&](size_t bytes) {
        void* r = ws + off;
        off = (off + bytes + 255) & ~(size_t)255;
        return r;
    };
    float* Xw     = (float*)alloc((size_t)nnodes * CCH * 4);  // 25.6 MB
    float* score  = (float*)alloc((size_t)nnodes * 4);
    float* scopy  = (float*)alloc((size_t)nnodes * 4);
    float* deg    = (float*)alloc((size_t)nnodes * 4);
    float* dinv   = (float*)alloc((size_t)nnodes * 4);
    float* xt     = (float*)alloc((size_t)CCH * CCH * 4);
    float* gi     = (float*)alloc((size_t)CCH * 3 * CCH * 4);
    float* gh     = (float*)alloc((size_t)CCH * 3 * CCH * 4);
    float* wnew   = (float*)alloc((size_t)CCH * CCH * 4);
    float* topv   = (float*)alloc((size_t)CCH * 4);
    int*   topi   = (int*)  alloc((size_t)CCH * 4);
    float* invnrm = (float*)alloc(4);
    float* out    = (float*)d_out;

    // ---- 1. TopK pooling ----
    pnorm_kernel<<<1, 32, 0, stream>>>(p, invnrm);
    {
        int waves = nnodes, wpb = 8;
        score_kernel<<<(waves + wpb - 1) / wpb, wpb * WAVE, 0, stream>>>(
            X, p, invnrm, score, scopy, nnodes);
    }
    topk_kernel<<<1, 1024, 0, stream>>>(scopy, topv, topi, nnodes);
    gather_gate_kernel<<<CCH, CCH, 0, stream>>>(X, topi, topv, xt);

    // ---- 2. GRU step: two small WMMA GEMMs + gates -> W_new ----
    {
        int tiles = (CCH / 16) * (3 * CCH / 16);     // 8*24 = 192
        wmma_gemm_nt_f32<<<tiles / 4, 4 * WAVE, 0, stream>>>(xt,   W_ih, gi, CCH, 3 * CCH, CCH);
        wmma_gemm_nt_f32<<<tiles / 4, 4 * WAVE, 0, stream>>>(W_cv, W_hh, gh, CCH, 3 * CCH, CCH);
    }
    gru_gate_kernel<<<(CCH * CCH) / 256, 256, 0, stream>>>(gi, gh, b_ih, b_hh, W_cv, wnew);

    // ---- 3. GCNConv ----
    // 3a. Xw = X @ W_new^T  (the big WMMA GEMM: 3125x8 = 25000 tiles)
    {
        int tiles = (nnodes / 16) * (CCH / 16);
        wmma_gemm_nt_f32<<<(tiles + 3) / 4, 4 * WAVE, 0, stream>>>(
            X, wnew, Xw, nnodes, CCH, CCH);
    }
    // 3b. degrees + symmetric norm
    zero_f32_kernel<<<(nnodes + 255) / 256, 256, 0, stream>>>(deg, nnodes);
    degree_kernel<<<(E + 255) / 256, 256, 0, stream>>>(ei, ew, deg, E);
    dinv_kernel<<<(nnodes + 255) / 256, 256, 0, stream>>>(deg, dinv, nnodes);
    // 3c. out = b_conv broadcast, then scatter-add normalized messages
    init_out_kernel<<<((size_t)nnodes * CCH + 255) / 256, 256, 0, stream>>>(
        out, b_cv, nnodes * CCH);
    {
        int total = E + nnodes, wpb = 8;
        scatter_kernel<<<(total + wpb - 1) / wpb, wpb * WAVE, 0, stream>>>(
            ei, ew, dinv, Xw, out, E, nnodes);
    }
}